// AdditiveAttention_20753281974975
// MI455X (gfx1250) — compile-verified
//
#include <hip/hip_runtime.h>
#include <hip/hip_bf16.h>

// Problem constants (match reference)
#define Bn 4
#define Qn 512
#define Kn 512
#define Dn 512   // DQ = DK = DV = 512
#define Hn 256

typedef __attribute__((ext_vector_type(2))) float v2f;
typedef __attribute__((ext_vector_type(8))) float v8f;

// ---------------------------------------------------------------------------
// CDNA5 transcendental helpers: single-instruction tanh / exp2 on the TRANS
// unit (quarter-rate, but vastly cheaper than the ocml polynomial path).
// ---------------------------------------------------------------------------
__device__ __forceinline__ float fast_tanh(float x) {
#if __has_builtin(__builtin_amdgcn_tanhf)
  return __builtin_amdgcn_tanhf(x);
#else
  float r;
  // v_nop covers the TRANS multicycle co-execution hazard (ISA 5.7.4).
  asm volatile("v_tanh_f32 %0, %1\n\tv_nop" : "=v"(r) : "v"(x));
  return r;
#endif
}

__device__ __forceinline__ float fast_exp2(float x) {
#if __has_builtin(__builtin_amdgcn_exp2f)
  return __builtin_amdgcn_exp2f(x);
#else
  float r;
  asm volatile("v_exp_f32 %0, %1\n\tv_nop" : "=v"(r) : "v"(x));
  return r;
#endif
}

// ---------------------------------------------------------------------------
// Wave-level f32 GEMM of one 16x16 output tile using V_WMMA_F32_16X16X4_F32.
//   C[r0:r0+16, c0:c0+16] = A[r0:r0+16, 0:kdim] * B[0:kdim, c0:c0+16]
// Fragment layouts per CDNA5 ISA 7.12.2 (32-bit A 16x4, B 4x16, C/D 16x16):
//   A: lanes 0-15 hold M=lane (K=+0,+1 in v0,v1); lanes 16-31 hold K=+2,+3
//   B: lanes 0-15 hold N=lane (K=+0,+1);          lanes 16-31 hold K=+2,+3
//   C: vgpr j -> M = j (lanes 0-15) / M = j+8 (lanes 16-31), N = lane&15
// ---------------------------------------------------------------------------
__device__ __forceinline__ void wave_gemm_16x16(
    const float* __restrict__ A, int lda,
    const float* __restrict__ Bm, int ldb,
    float* __restrict__ C, int ldc,
    int kdim, int r0, int c0, int lane)
{
  const int m     = lane & 15;
  const int khalf = (lane >> 4) << 1;            // 0 or 2
  const float* ap = A  + (size_t)(r0 + m) * lda + khalf;
  const float* bp = Bm + (size_t)khalf * ldb + c0 + m;

  v8f c = {0.f, 0.f, 0.f, 0.f, 0.f, 0.f, 0.f, 0.f};
#pragma unroll 4
  for (int d = 0; d < kdim; d += 4) {
    v2f a, b;
    a.x = ap[d];                                 // contiguous pair -> b64 load
    a.y = ap[d + 1];
    b.x = bp[(size_t)d * ldb];
    b.y = bp[(size_t)(d + 1) * ldb];
    c = __builtin_amdgcn_wmma_f32_16x16x4_f32(
        /*neg_a=*/false, a, /*neg_b=*/false, b,
        /*c_mod=*/(short)0, c, /*reuse_a=*/false, /*reuse_b=*/false);
  }

  const int rbase = r0 + ((lane >> 4) << 3);     // +8 rows for high half-wave
#pragma unroll
  for (int j = 0; j < 8; ++j)
    C[(size_t)(rbase + j) * ldc + c0 + m] = c[j];
}

// ---------------------------------------------------------------------------
// Kernel 1: both linear projections (queries*W_q and keys*W_k) -> workspace.
// 4096 wave-tiles total (2 mats * 128 row-tiles * 16 h-tiles), 8 waves/block.
// ---------------------------------------------------------------------------
__global__ __launch_bounds__(256) void proj_kernel(
    const float* __restrict__ queries, const float* __restrict__ keys,
    const float* __restrict__ Wq, const float* __restrict__ Wk,
    float* __restrict__ qproj, float* __restrict__ kproj)
{
  const int lane = threadIdx.x & 31;
  const int wid  = threadIdx.x >> 5;
  int tile = blockIdx.x * 8 + wid;               // 0..4095
  const int whichK = (tile >= 2048);
  tile &= 2047;
  const int rowTile = tile >> 4;                 // 0..127  (B*Q/16 rows)
  const int hTile   = tile & 15;                 // 0..15   (H/16 cols)

  const float* X = whichK ? keys : queries;
  const float* W = whichK ? Wk : Wq;
  float*       P = whichK ? kproj : qproj;

  wave_gemm_16x16(X, Dn, W, Hn, P, Hn, Dn, rowTile * 16, hTile * 16, lane);
}

// ---------------------------------------------------------------------------
// Kernel 2: additive scores  s[b,q,k] = sum_h w_v[h]*tanh(qp[b,q,h]+kp[b,k,h])
// One block computes a 16x16 (q,k) tile; thread t -> (q=t>>4, k=t&15).
// LDS tiles padded to stride 260 floats: 260 % 64 == 4 banks -> the 16 lanes
// with distinct k hit 16 distinct banks (conflict-free), and 260*4 bytes is
// 16B-aligned so rows can be read as float4 (ds_load_b128).
// ---------------------------------------------------------------------------
__global__ __launch_bounds__(256) void score_kernel(
    const float* __restrict__ qproj, const float* __restrict__ kproj,
    const float* __restrict__ wv, float* __restrict__ scores)
{
  __shared__ __align__(16) float qpL[16 * 260];
  __shared__ __align__(16) float kpL[16 * 260];
  __shared__ __align__(16) float wvL[Hn];

  const int t  = threadIdx.x;
  int bid = blockIdx.x;                          // b*1024 + qt*32 + kt
  const int kt = bid & 31; bid >>= 5;
  const int qt = bid & 31; bid >>= 5;
  const int b  = bid;

  const int qrow0 = b * Qn + qt * 16;
  const int krow0 = b * Kn + kt * 16;

  wvL[t] = wv[t];                                // H == blockDim == 256
#pragma unroll
  for (int pass = 0; pass < 16; ++pass) {        // 16*256 elems, 256 threads
    const int idx = pass * 256 + t;
    const int r = idx >> 8, h = idx & 255;       // coalesced in h
    qpL[r * 260 + h] = qproj[(size_t)(qrow0 + r) * Hn + h];
    kpL[r * 260 + h] = kproj[(size_t)(krow0 + r) * Hn + h];
  }
  __syncthreads();

  const int ql = t >> 4, kl = t & 15;
  const float4* qv  = reinterpret_cast<const float4*>(qpL + ql * 260);
  const float4* kv  = reinterpret_cast<const float4*>(kpL + kl * 260);
  const float4* wvv = reinterpret_cast<const float4*>(wvL);

  float acc = 0.f;
#pragma unroll 8
  for (int h4 = 0; h4 < Hn / 4; ++h4) {
    const float4 qh = qv[h4];
    const float4 kh = kv[h4];
    const float4 wh = wvv[h4];
    acc += wh.x * fast_tanh(qh.x + kh.x);
    acc += wh.y * fast_tanh(qh.y + kh.y);
    acc += wh.z * fast_tanh(qh.z + kh.z);
    acc += wh.w * fast_tanh(qh.w + kh.w);
  }

  scores[(size_t)(qrow0 + ql) * Kn + kt * 16 + kl] = acc;
}

// ---------------------------------------------------------------------------
// Kernel 3: row softmax over K=512 (one block per (b,q) row), in-place.
// ---------------------------------------------------------------------------
__global__ __launch_bounds__(256) void softmax_kernel(float* __restrict__ scores)
{
  __shared__ float red[256];
  const int t = threadIdx.x;
  float* row = scores + (size_t)blockIdx.x * Kn;

  const float a = row[t];
  const float b = row[t + 256];

  red[t] = fmaxf(a, b);
  __syncthreads();
#pragma unroll
  for (int s = 128; s > 0; s >>= 1) {
    if (t < s) red[t] = fmaxf(red[t], red[t + s]);
    __syncthreads();
  }
  const float m = red[0];
  __syncthreads();

  const float L2E = 1.4426950408889634f;
  const float ea = fast_exp2((a - m) * L2E);
  const float eb = fast_exp2((b - m) * L2E);

  red[t] = ea + eb;
  __syncthreads();
#pragma unroll
  for (int s = 128; s > 0; s >>= 1) {
    if (t < s) red[t] += red[t + s];
    __syncthreads();
  }
  const float inv = 1.f / red[0];

  row[t]       = ea * inv;
  row[t + 256] = eb * inv;
}

// ---------------------------------------------------------------------------
// Kernel 4: out[b,q,:] = attn[b,q,:] * values[b]  (f32 WMMA wave-GEMM)
// 4096 tiles = 128 row-tiles (B*Q/16) * 32 v-tiles (DV/16), 8 waves/block.
// ---------------------------------------------------------------------------
__global__ __launch_bounds__(256) void out_kernel(
    const float* __restrict__ attn, const float* __restrict__ values,
    float* __restrict__ out)
{
  const int lane = threadIdx.x & 31;
  const int wid  = threadIdx.x >> 5;
  const int tile = blockIdx.x * 8 + wid;         // 0..4095
  const int rowTile = tile >> 5;                 // 0..127
  const int vTile   = tile & 31;                 // 0..31
  const int r0 = rowTile * 16;
  const int b  = r0 >> 9;                        // 16-row tile never crosses b

  wave_gemm_16x16(attn, Kn,
                  values + (size_t)b * Kn * Dn, Dn,
                  out, Dn,
                  Kn, r0, vTile * 16, lane);
}

// ---------------------------------------------------------------------------
// Host entry
// ---------------------------------------------------------------------------
extern "C" void kernel_launch(void* const* d_in, const int* in_sizes, int n_in,
                              void* d_out, int out_size, void* d_ws, size_t ws_size,
                              hipStream_t stream) {
  const float* queries = (const float*)d_in[0];  // [B,Q,D]
  const float* keys    = (const float*)d_in[1];  // [B,K,D]
  const float* values  = (const float*)d_in[2];  // [B,K,D]
  const float* Wq      = (const float*)d_in[3];  // [D,H]
  const float* Wk      = (const float*)d_in[4];  // [D,H]
  const float* wv      = (const float*)d_in[5];  // [H]
  float* out = (float*)d_out;                    // [B,Q,D]

  // Workspace layout: qproj(2MB) | kproj(2MB) | scores/attn(4MB) = 8MB
  float* qproj  = (float*)d_ws;                  // [B*Q, H]
  float* kproj  = qproj + (size_t)Bn * Qn * Hn;  // [B*K, H]
  float* scores = kproj + (size_t)Bn * Kn * Hn;  // [B*Q, K]

  proj_kernel   <<<512,  256, 0, stream>>>(queries, keys, Wq, Wk, qproj, kproj);
  score_kernel  <<<Bn * 32 * 32, 256, 0, stream>>>(qproj, kproj, wv, scores);
  softmax_kernel<<<Bn * Qn, 256, 0, stream>>>(scores);
  out_kernel    <<<512,  256, 0, stream>>>(scores, values, out);
}